// SSIMLoss_48112223650121
// MI455X (gfx1250) — compile-verified
//
#include <hip/hip_runtime.h>
#include <hip/hip_bf16.h>

typedef float v2f __attribute__((ext_vector_type(2)));
typedef float v8f __attribute__((ext_vector_type(8)));

#define IMG_W 512
#define PATCHES_PER_PLANE 1024      /* (512/16)^2 */
#define PLANE_ELEMS (512 * 512)
#define GRID_MAIN 512
#define BLOCK_MAIN 256
#define WAVES_PER_BLOCK 8
#define C1_SSIM 1.0e-4f             /* (0.01*1.0)^2 */
#define C2_SSIM 9.0e-4f             /* (0.03*1.0)^2 */
#define U_STRIDE 17                 /* padded row stride for LDS transpose staging */
#define WAVE_LDS 800                /* floats per wave: X 256 | Y 256 | U 272 (+pad) */

#if defined(__gfx1250__) && __has_builtin(__builtin_amdgcn_global_load_async_to_lds_b128)
#define USE_ASYNC_LDS 1
#else
#define USE_ASYNC_LDS 0
#endif

#if USE_ASYNC_LDS
// parameter type per clang diagnostic: 'vector_size(4*sizeof(int)) int __device__ *'
typedef int v4i __attribute__((vector_size(4 * sizeof(int))));
typedef __attribute__((address_space(1))) v4i gv4;   // global (__device__) 128-bit
typedef __attribute__((address_space(3))) v4i lv4;   // LDS (__shared__) 128-bit
__device__ __forceinline__ void wait_asynccnt0() {
#if __has_builtin(__builtin_amdgcn_s_wait_asynccnt)
  __builtin_amdgcn_s_wait_asynccnt(0);
  asm volatile("" ::: "memory");
#else
  asm volatile("s_wait_asynccnt 0x0" ::: "memory");
#endif
}
#endif

// W16(m,k): block-diagonal 16x16 "pad+gauss" matrix, W = A(8x18 toeplitz gauss) * R(18x8 reflect)
__device__ __forceinline__ float wval16(const float* g, int m, int k) {
  if (((m ^ k) & 8) != 0) return 0.0f;      // off-diagonal 8x8 block -> 0
  int mm = m & 7, kk = k & 7;
  float s = 0.0f;
#pragma unroll
  for (int t = 0; t < 11; ++t) {
    int i = mm + t - 5;                      // unpadded coordinate (may be out of [0,8))
    int r = (i < 0) ? -i : ((i > 7) ? 14 - i : i);   // numpy/jnp 'reflect'
    s += (r == kk) ? g[t] : 0.0f;
  }
  return s;
}

template <int S>
__device__ __forceinline__ float statf(float a, float b) {
  if (S == 0) return a;
  if (S == 1) return b;
  if (S == 2) return a * a;
  if (S == 3) return b * b;
  return a * b;
}

// O = W16 * T * W16^T for one 16x16 patch, via two chained 16x16x16 GEMMs
// (each = 4 x V_WMMA_F32_16X16X4_F32). U is re-laid-out D->A through LDS.
template <int S>
__device__ __forceinline__ v8f conv_stat(const float* xr, const float* yr,
                                         const v2f* wa, float* wU,
                                         int n16, int koff) {
  v8f u = {};
#pragma unroll
  for (int c = 0; c < 4; ++c) {
    v2f b;
    b.x = statf<S>(xr[2 * c + 0], yr[2 * c + 0]);
    b.y = statf<S>(xr[2 * c + 1], yr[2 * c + 1]);
    u = __builtin_amdgcn_wmma_f32_16x16x4_f32(false, wa[c], false, b,
                                              (short)0, u, false, false);
  }
  int mbase = (koff >> 1) << 3;              // laneHi * 8
#pragma unroll
  for (int v = 0; v < 8; ++v)                // stage U (D-layout rows)
    wU[(mbase + v) * U_STRIDE + n16] = u[v];
  v8f o = {};
#pragma unroll
  for (int c = 0; c < 4; ++c) {              // A-operand: U(m=n16, k=4c+v+koff)
    v2f a2;
    a2.x = wU[n16 * U_STRIDE + 4 * c + 0 + koff];
    a2.y = wU[n16 * U_STRIDE + 4 * c + 1 + koff];
    o = __builtin_amdgcn_wmma_f32_16x16x4_f32(false, a2, false, wa[c],
                                              (short)0, o, false, false);
  }
  return o;
}

__global__ __launch_bounds__(BLOCK_MAIN) void ssim_wmma_kernel(
    const float* __restrict__ X, const float* __restrict__ Y,
    float* __restrict__ partial, int npatch) {
  __shared__ float smem[WAVES_PER_BLOCK * WAVE_LDS + BLOCK_MAIN];
  const int tid = threadIdx.x;
  const int lane = tid & 31;
  const int waveId = tid >> 5;
  const int laneHi = (lane >> 4) & 1;
  const int n16 = lane & 15;
  const int koff = 2 * laneHi;               // K offset of this lane-half (32-bit operands)

  // normalized 1D gaussian, sigma = 1.5, ksize = 11
  float g[11];
  float gs = 0.0f;
#pragma unroll
  for (int t = 0; t < 11; ++t) {
    float c = (float)t - 5.0f;
    g[t] = expf(-c * c / 4.5f);
    gs += g[t];
  }
  const float ginv = 1.0f / gs;
#pragma unroll
  for (int t = 0; t < 11; ++t) g[t] *= ginv;

  // per-lane constant WMMA operand: wa[c][v] = W16(n16, 4c+v+koff)
  // (serves as A of GEMM1 and as B of GEMM2, since B(k,n)=W16^T(k,n)=W16(n,k))
  v2f wa[4];
#pragma unroll
  for (int c = 0; c < 4; ++c) {
    wa[c].x = wval16(g, n16, 4 * c + 0 + koff);
    wa[c].y = wval16(g, n16, 4 * c + 1 + koff);
  }

  float* wX = smem + waveId * WAVE_LDS;      // 256 floats
  float* wY = wX + 256;                      // 256 floats
  float* wU = wX + 512;                      // 272 floats, disjoint from X/Y tiles

  // per-lane tile coordinates for the 16x16 loads (2 x float4 per array)
  const int r0 = lane >> 2;                  // rows 0..7 (i=0), +8 for i=1
  const int c40 = (lane & 3) << 2;

  float acc = 0.0f;
  const int gw = blockIdx.x * WAVES_PER_BLOCK + waveId;
  const int nwaves = gridDim.x * WAVES_PER_BLOCK;

  for (int P = gw; P < npatch; P += nwaves) {
    const int plane = P >> 10;
    const int pp = P & 1023;
    const int pr = pp >> 5;
    const int pc = pp & 31;
    const size_t base = (size_t)plane * PLANE_ELEMS + (size_t)(pr * 16) * IMG_W + pc * 16;
    const float* xb = X + base;
    const float* yb = Y + base;

#if USE_ASYNC_LDS
    {
      // rows 0..7 via (r0,c40); rows 8..15 via (r0+8,c40)
      __builtin_amdgcn_global_load_async_to_lds_b128(
          (gv4*)(xb + r0 * IMG_W + c40), (lv4*)(wX + r0 * 16 + c40), 0, 0);
      __builtin_amdgcn_global_load_async_to_lds_b128(
          (gv4*)(xb + (r0 + 8) * IMG_W + c40), (lv4*)(wX + (r0 + 8) * 16 + c40), 0, 0);
      __builtin_amdgcn_global_load_async_to_lds_b128(
          (gv4*)(yb + r0 * IMG_W + c40), (lv4*)(wY + r0 * 16 + c40), 0, 0);
      __builtin_amdgcn_global_load_async_to_lds_b128(
          (gv4*)(yb + (r0 + 8) * IMG_W + c40), (lv4*)(wY + (r0 + 8) * 16 + c40), 0, 0);
      wait_asynccnt0();
    }
#else
#pragma unroll
    for (int i = 0; i < 2; ++i) {            // 16x16 tile -> LDS (float4 lanes)
      int r = r0 + 8 * i;
      float4 vx = *(const float4*)(xb + r * IMG_W + c40);
      float4 vy = *(const float4*)(yb + r * IMG_W + c40);
      *(float4*)(wX + r * 16 + c40) = vx;
      *(float4*)(wY + r * 16 + c40) = vy;
    }
#endif

    // B-operand columns for all K-chunks, pulled once for all 5 stats
    float xr[8], yr[8];
#pragma unroll
    for (int c = 0; c < 4; ++c)
#pragma unroll
      for (int v = 0; v < 2; ++v) {
        int k = 4 * c + v + koff;
        xr[2 * c + v] = wX[k * 16 + n16];
        yr[2 * c + v] = wY[k * 16 + n16];
      }

    v8f o0 = conv_stat<0>(xr, yr, wa, wU, n16, koff);  // mu_x
    v8f o1 = conv_stat<1>(xr, yr, wa, wU, n16, koff);  // mu_y
    v8f o2 = conv_stat<2>(xr, yr, wa, wU, n16, koff);  // mu_xx
    v8f o3 = conv_stat<3>(xr, yr, wa, wU, n16, koff);  // mu_yy
    v8f o4 = conv_stat<4>(xr, yr, wa, wU, n16, koff);  // mu_xy

#pragma unroll
    for (int v = 0; v < 8; ++v) {
      float mx = o0[v], my = o1[v];
      float mxx = o2[v], myy = o3[v], mxy = o4[v];
      float sx = mxx - mx * mx;
      float sy = myy - my * my;
      float sxy = mxy - mx * my;
      float num = (2.0f * mx * my + C1_SSIM) * (2.0f * sxy + C2_SSIM);
      float den = (mx * mx + my * my + C1_SSIM) * (sx + sy + C2_SSIM);
      acc += num / den;
    }
  }

  // deterministic in-block tree reduction
  float* red = smem + WAVES_PER_BLOCK * WAVE_LDS;
  red[tid] = acc;
  __syncthreads();
#pragma unroll
  for (int s = BLOCK_MAIN / 2; s > 0; s >>= 1) {
    if (tid < s) red[tid] += red[tid + s];
    __syncthreads();
  }
  if (tid == 0) partial[blockIdx.x] = red[0];
}

__global__ __launch_bounds__(256) void ssim_reduce_kernel(
    const float* __restrict__ partial, int nPart, float* __restrict__ out,
    long long nElems) {
  __shared__ double sd[256];
  const int tid = threadIdx.x;
  double s = 0.0;
  for (int i = tid; i < nPart; i += 256) s += (double)partial[i];
  sd[tid] = s;
  __syncthreads();
  for (int k = 128; k > 0; k >>= 1) {
    if (tid < k) sd[tid] += sd[tid + k];
    __syncthreads();
  }
  if (tid == 0) out[0] = (float)(1.0 - sd[0] / (double)nElems);
}

extern "C" void kernel_launch(void* const* d_in, const int* in_sizes, int n_in,
                              void* d_out, int out_size, void* d_ws, size_t ws_size,
                              hipStream_t stream) {
  (void)n_in; (void)out_size; (void)ws_size;
  const float* X = (const float*)d_in[0];   // inputs
  const float* Y = (const float*)d_in[1];   // targets (SSIM symmetric in x/y)
  float* out = (float*)d_out;
  float* partial = (float*)d_ws;            // GRID_MAIN floats of scratch

  long long n = (long long)in_sizes[0];     // 16*3*512*512
  int planes = (int)(n / (long long)PLANE_ELEMS);
  int npatch = planes * PATCHES_PER_PLANE;

  ssim_wmma_kernel<<<GRID_MAIN, BLOCK_MAIN, 0, stream>>>(X, Y, partial, npatch);
  ssim_reduce_kernel<<<1, 256, 0, stream>>>(partial, GRID_MAIN, out, n);
}